// ContextAwareArtRecSys_33389075759122
// MI455X (gfx1250) — compile-verified
//
#include <hip/hip_runtime.h>

typedef __attribute__((ext_vector_type(2))) float v2f;
typedef __attribute__((ext_vector_type(4))) float v4f;
typedef __attribute__((ext_vector_type(8))) float v8f;

#define N_USERS 50000
#define N_ITEMS 100000
#define N_EDGES 500000
#define HDIM    256
#define HALF    128

// ---------------------------------------------------------------------------
// Kernel A: fold w_out into the two projections (tiny: one block).
//   u_vec[h] = sum_j w_out[j]      * w_user[j,h]
//   i_vec[h] = sum_j w_out[128+j]  * w_item[j,h]
//   c        = w_out[:128].b_user + w_out[128:].b_item + b_out
// ---------------------------------------------------------------------------
__global__ void fold_kernel(const float* __restrict__ w_user, const float* __restrict__ b_user,
                            const float* __restrict__ w_item, const float* __restrict__ b_item,
                            const float* __restrict__ w_out,  const float* __restrict__ b_out,
                            float* __restrict__ ws) {
    const int h = threadIdx.x;          // 0..255
    float us = 0.0f, is = 0.0f;
    #pragma unroll 4
    for (int j = 0; j < HALF; ++j) {
        us = fmaf(w_out[j],        w_user[j * HDIM + h], us);
        is = fmaf(w_out[HALF + j], w_item[j * HDIM + h], is);
    }
    ws[h]        = us;   // u_vec
    ws[HDIM + h] = is;   // i_vec
    if (h == 0) {
        float c = b_out[0];
        for (int j = 0; j < HALF; ++j) {
            c = fmaf(w_out[j], b_user[j], c);
            c = fmaf(w_out[HALF + j], b_item[j], c);
        }
        ws[2 * HDIM] = c;
    }
}

// ---------------------------------------------------------------------------
// Kernel B: dual GEMV  s[0:50000]   = z_user @ u_vec
//                      s[50000: ]   = z_item @ i_vec
// One wave32 handles 16 rows via V_WMMA_F32_16X16X4_F32, K stepped by 4.
// A-layout (ISA 7.12.2): lane m=lane&15; lanes>=16 hold K+2,K+3.
// B is the folded vector broadcast across all 16 columns, so every column of
// D holds the same per-row dot product.
// ---------------------------------------------------------------------------
__global__ void __launch_bounds__(256)
gemv_wmma_kernel(const float* __restrict__ z_user, const float* __restrict__ z_item,
                 const float* __restrict__ ws, float* __restrict__ s) {
    const int lane = threadIdx.x & 31;
    const int wid  = blockIdx.x * (blockDim.x >> 5) + (threadIdx.x >> 5);
    const int user_waves  = N_USERS / 16;                   // 3125 (exact)
    const int total_waves = (N_USERS + N_ITEMS) / 16;       // 9375 (exact)
    if (wid >= total_waves) return;                         // uniform per wave

    const bool  is_item = (wid >= user_waves);
    const float* __restrict__ z   = is_item ? z_item : z_user;
    const float* __restrict__ vec = ws + (is_item ? HDIM : 0);
    const int row_base = (is_item ? (wid - user_waves) : wid) * 16;

    const int m     = lane & 15;
    const int khalf = (lane >> 4) * 2;                      // 0 or 2
    const float* __restrict__ arow = z + (size_t)(row_base + m) * HDIM + khalf;
    const float* __restrict__ brow = vec + khalf;

    v8f acc = {};
    #pragma unroll 8
    for (int k = 0; k < HDIM; k += 4) {
        v2f a = *(const v2f*)(arow + k);                    // A[m, k+khalf .. +1]
        v2f b = *(const v2f*)(brow + k);                    // B[k+khalf .. +1] bcast over N
        acc = __builtin_amdgcn_wmma_f32_16x16x4_f32(
            /*neg_a=*/false, a, /*neg_b=*/false, b,
            /*c_mod=*/(short)0, acc, /*reuse_a=*/false, /*reuse_b=*/false);
    }

    // D layout: VGPR r, lanes 0-15 -> M=r; lanes 16-31 -> M=r+8 (value same for all N).
    if (m == 0) {
        float* out = s + (is_item ? N_USERS : 0) + row_base + (lane >> 4) * 8;
        const v4f* ap = (const v4f*)&acc;
        *(v4f*)(out)     = ap[0];
        *(v4f*)(out + 4) = ap[1];
    }
}

// ---------------------------------------------------------------------------
// Kernel C: per-edge combine  out[e] = s_u[src[e]] + s_i[dst[e]] + c
// Score table (~600 KB) is L2-resident; streams 8 MB of indices, 2 MB out.
// ---------------------------------------------------------------------------
__global__ void edge_kernel(const long long* __restrict__ esrc, const long long* __restrict__ edst,
                            const float* __restrict__ s, const float* __restrict__ cptr,
                            float* __restrict__ out, int n) {
    const int e = blockIdx.x * blockDim.x + threadIdx.x;
    if (e >= n) return;
    const float c = cptr[0];
    out[e] = s[(int)esrc[e]] + s[N_USERS + (int)edst[e]] + c;
}

extern "C" void kernel_launch(void* const* d_in, const int* in_sizes, int n_in,
                              void* d_out, int out_size, void* d_ws, size_t ws_size,
                              hipStream_t stream) {
    const float*     z_user = (const float*)d_in[0];
    const float*     z_item = (const float*)d_in[1];
    const long long* esrc   = (const long long*)d_in[2];
    const long long* edst   = (const long long*)d_in[3];
    const float*     w_user = (const float*)d_in[4];
    const float*     b_user = (const float*)d_in[5];
    const float*     w_item = (const float*)d_in[6];
    const float*     b_item = (const float*)d_in[7];
    const float*     w_out  = (const float*)d_in[8];
    const float*     b_out  = (const float*)d_in[9];

    float* ws  = (float*)d_ws;
    float* out = (float*)d_out;
    float* svec = ws + 576;   // 16-float aligned score table, 150000 floats

    fold_kernel<<<1, 256, 0, stream>>>(w_user, b_user, w_item, b_item, w_out, b_out, ws);

    const int total_waves = (N_USERS + N_ITEMS) / 16;       // 9375
    const int waves_per_block = 8;                          // 256 threads
    const int blocks = (total_waves + waves_per_block - 1) / waves_per_block;
    gemv_wmma_kernel<<<blocks, waves_per_block * 32, 0, stream>>>(z_user, z_item, ws, svec);

    edge_kernel<<<(N_EDGES + 255) / 256, 256, 0, stream>>>(esrc, edst, svec, ws + 512, out, N_EDGES);
}